// IntegrateColorAndWeightsModule_10033043603890
// MI455X (gfx1250) — compile-verified
//
#include <hip/hip_runtime.h>
#include <stdint.h>

// ---------------------------------------------------------------------------
// Segmented reduction (ray color integration) for MI455X / gfx1250.
//   pred_rgb[r] = sum_{i : ray_indices[i]==r} weights[i] * rgb[i]
//
// Memory-bound: 80 MB read @ 23.3 TB/s ~= 3.4 us floor. This version:
//   * 4 contiguous samples per lane (wave-tile = 128 samples) so the wave32
//     segmented scan is amortized 4x -> ~0.8 instr/sample, below HBM floor.
//   * rgb (48 MB stream) staged via gfx1250 async DMA
//     global_load_async_to_lds_b128 + ASYNCcnt/s_wait_asynccnt,
//     per-wave private double buffer (no block barriers).
//   * idx/w: coalesced 16B nontemporal loads (native clang vector types).
//   * sorted keys -> per-lane run merge + wave segmented scan + carry into
//     the next lane's first run; segment tails flush with f32 atomics.
// ---------------------------------------------------------------------------

#define TPB 256
#define WAVES_PER_BLOCK (TPB / 32)
#define TILE 128                         // samples per wave-tile (4 per lane)
#define TILE_F4 (TILE * 12 / 16)         // 96 float4 chunks per tile buffer
#define CHUNKS_PER_LANE 3                // 96 / 32

// Native clang vectors: accepted by __builtin_nontemporal_load (HIP's
// int4/float4 structs are not).
typedef int   vi4 __attribute__((ext_vector_type(4)));
typedef float vf4 __attribute__((ext_vector_type(4)));

__global__ __launch_bounds__(TPB) void zero_out_kernel(float* __restrict__ out, int n) {
    int i = blockIdx.x * blockDim.x + threadIdx.x;
    if (i < n) out[i] = 0.0f;
}

__device__ __forceinline__ void flush3(float* __restrict__ out, int key,
                                       float x, float y, float z) {
    if (key >= 0) {
        atomicAdd(&out[(int64_t)key * 3 + 0], x);
        atomicAdd(&out[(int64_t)key * 3 + 1], y);
        atomicAdd(&out[(int64_t)key * 3 + 2], z);
    }
}

// Stage one 128-sample rgb tile (1536 B) into LDS: 96 x b128 chunks, three
// async instructions with all 32 lanes active. Per-lane clamped addresses keep
// every issue valid (EXEC != 0) so ASYNCcnt increments exactly 3x per call.
__device__ __forceinline__ void stage_rgb_tile(const float* __restrict__ rgb,
                                               int64_t n_samples,
                                               int64_t first_sample,
                                               uint32_t lds_byte_off,
                                               int lane) {
    const int64_t bmax = ((int64_t)n_samples * 12 - 16) & ~(int64_t)15;
#pragma unroll
    for (int j = 0; j < CHUNKS_PER_LANE; ++j) {
        int chunk = lane + 32 * j;
        int64_t b = first_sample * 12 + (int64_t)chunk * 16;
        if (b > bmax) b = bmax;
        if (b < 0) b = 0;
        const char* gptr = (const char*)rgb + b;
        uint32_t lptr = lds_byte_off + (uint32_t)chunk * 16u;
        asm volatile("global_load_async_to_lds_b128 %0, %1, off"
                     :
                     : "v"(lptr), "v"(gptr)
                     : "memory");
    }
}

__global__ __launch_bounds__(TPB) void integrate_rays_kernel(
    const float* __restrict__ rgb,      // [n, 3] f32
    const float* __restrict__ w,        // [n] f32
    const int*   __restrict__ idx,      // [n] i32, sorted non-decreasing
    float*       __restrict__ out,      // [n_rays, 3] f32, pre-zeroed
    int n, int iters) {

    __shared__ float4 lds_rgb[WAVES_PER_BLOCK][2][TILE_F4]; // 24 KB / block

    const int lane = threadIdx.x & 31;
    const int wid  = threadIdx.x >> 5;
    const int64_t pass = (int64_t)gridDim.x * TPB * 4;       // samples per grid pass
    const int64_t s0 = ((int64_t)blockIdx.x * WAVES_PER_BLOCK + wid) * TILE;

    const uint32_t lbuf0 = (uint32_t)(uintptr_t)&lds_rgb[wid][0][0];
    const uint32_t lbuf1 = (uint32_t)(uintptr_t)&lds_rgb[wid][1][0];

    // Prologue: stage tile 0 into buffer 0.
    stage_rgb_tile(rgb, n, s0, lbuf0, lane);

    for (int t = 0; t < iters; ++t) {
        // Prefetch tile t+1 into the other buffer (clamped-safe past the end).
        stage_rgb_tile(rgb, n, s0 + (int64_t)(t + 1) * pass,
                       (t & 1) ? lbuf0 : lbuf1, lane);

        // Outstanding async instrs per wave <= 6; wait until <= 3 -> tile t
        // fully landed (per-wave in-order completion).
        asm volatile("s_wait_asynccnt 0x3" ::: "memory");

        const int64_t tile_base = s0 + (int64_t)t * pass;
        const int64_t g0 = tile_base + 4 * lane;

        // ---- load 4 samples: keys, weights, rgb (from staged LDS) ----------
        int   k0, k1, k2, k3;
        float v0x, v0y, v0z, v1x, v1y, v1z, v2x, v2y, v2z, v3x, v3y, v3z;

        const float4* lb = &lds_rgb[wid][t & 1][3 * lane];
        float4 r0 = lb[0], r1 = lb[1], r2 = lb[2];

        if (tile_base + TILE <= (int64_t)n) {
            // Fast path: whole wave-tile in range (the only path when n is a
            // multiple of the pass size, as here: n = 2^22).
            vi4 kk = __builtin_nontemporal_load((const vi4*)(idx + g0));
            vf4 ww = __builtin_nontemporal_load((const vf4*)(w + g0));
            k0 = kk.x; k1 = kk.y; k2 = kk.z; k3 = kk.w;
            v0x = ww.x * r0.x; v0y = ww.x * r0.y; v0z = ww.x * r0.z;
            v1x = ww.y * r0.w; v1y = ww.y * r1.x; v1z = ww.y * r1.y;
            v2x = ww.z * r1.z; v2y = ww.z * r1.w; v2z = ww.z * r2.x;
            v3x = ww.w * r2.y; v3y = ww.w * r2.z; v3z = ww.w * r2.w;
        } else {
            // Tail path: per-sample guards; invalid samples get key=-1, val=0.
            float rr[12] = {r0.x, r0.y, r0.z, r0.w, r1.x, r1.y,
                            r1.z, r1.w, r2.x, r2.y, r2.z, r2.w};
            int   kt[4];
            float vt[4][3];
#pragma unroll
            for (int j = 0; j < 4; ++j) {
                int64_t g = g0 + j;
                if (g < (int64_t)n) {
                    kt[j] = idx[g];
                    float wv = w[g];
                    vt[j][0] = wv * rr[3 * j + 0];
                    vt[j][1] = wv * rr[3 * j + 1];
                    vt[j][2] = wv * rr[3 * j + 2];
                } else {
                    kt[j] = -1;
                    vt[j][0] = vt[j][1] = vt[j][2] = 0.0f;
                }
            }
            k0 = kt[0]; k1 = kt[1]; k2 = kt[2]; k3 = kt[3];
            v0x = vt[0][0]; v0y = vt[0][1]; v0z = vt[0][2];
            v1x = vt[1][0]; v1y = vt[1][1]; v1z = vt[1][2];
            v2x = vt[2][0]; v2y = vt[2][1]; v2z = vt[2][2];
            v3x = vt[3][0]; v3y = vt[3][1]; v3z = vt[3][2];
        }

        // ---- per-lane serial run merge over 4 sorted samples ---------------
        // First run is held back (it may receive a carry from the previous
        // lane); interior runs flush immediately; last run enters the scan.
        const int firstk = k0;
        int   curk = k0;
        float ax = v0x, ay = v0y, az = v0z;
        float fx = 0.f, fy = 0.f, fz = 0.f;
        bool haveFirst = false;

#define RUN_STEP(KJ, VX, VY, VZ)                                           \
        if ((KJ) == curk) { ax += (VX); ay += (VY); az += (VZ); }          \
        else {                                                             \
            if (!haveFirst) { haveFirst = true; fx = ax; fy = ay; fz = az; } \
            else            { flush3(out, curk, ax, ay, az); }             \
            curk = (KJ); ax = (VX); ay = (VY); az = (VZ);                  \
        }
        RUN_STEP(k1, v1x, v1y, v1z)
        RUN_STEP(k2, v2x, v2y, v2z)
        RUN_STEP(k3, v3x, v3y, v3z)
#undef RUN_STEP
        const int lastk = curk;

        // ---- wave32 segmented inclusive scan over last-run partials --------
        // Keys sorted => equality of lastk across a lane range implies the
        // whole range is one segment; plain compare-scan is exact.
#pragma unroll
        for (int off = 1; off < 32; off <<= 1) {
            int   pk = __shfl_up(lastk, off, 32);
            float px = __shfl_up(ax,   off, 32);
            float py = __shfl_up(ay,   off, 32);
            float pz = __shfl_up(az,   off, 32);
            if (lane >= off && pk == lastk) { ax += px; ay += py; az += pz; }
        }

        // ---- carry into this lane's first run, then flush it ---------------
        // (scan path and carry path are mutually exclusive: scan fires iff
        // prev.lastk == lastk; carry fires iff firstk != lastk.)
        int   ck = __shfl_up(lastk, 1, 32);
        float cx = __shfl_up(ax, 1, 32);
        float cy = __shfl_up(ay, 1, 32);
        float cz = __shfl_up(az, 1, 32);
        if (haveFirst) {
            if (lane > 0 && ck == firstk) { fx += cx; fy += cy; fz += cz; }
            flush3(out, firstk, fx, fy, fz);
        }

        // ---- segment-tail flush of the scanned last run --------------------
        int nfk = __shfl_down(firstk, 1, 32);   // next lane's FIRST key
        bool tail = (lane == 31) || (nfk != lastk);
        if (tail) flush3(out, lastk, ax, ay, az);
    }
}

extern "C" void kernel_launch(void* const* d_in, const int* in_sizes, int n_in,
                              void* d_out, int out_size, void* d_ws, size_t ws_size,
                              hipStream_t stream) {
    const float* rgb = (const float*)d_in[0];   // [n, 3] f32
    const float* w   = (const float*)d_in[1];   // [n, 1] f32
    const int*   idx = (const int*)d_in[2];     // [n] i32 (sorted)
    // d_in[3] = n_rays scalar; out_size == n_rays * 3 already.
    float* out = (float*)d_out;

    const int n = in_sizes[1]; // N_SAMPLES (weights element count)

    // Zero the output (harness poisons it; atomics accumulate into it).
    int zblocks = (out_size + TPB - 1) / TPB;
    if (zblocks > 0)
        zero_out_kernel<<<zblocks, TPB, 0, stream>>>(out, out_size);

    // 1024 blocks x 256 threads x 4 samples/thread = 2^20 samples per pass;
    // n = 2^22 -> 4 uniform iterations, tails handled by the guarded path.
    const int nblocks = 1024;
    const int64_t pass = (int64_t)nblocks * TPB * 4;
    const int iters = (int)((n + pass - 1) / pass);

    integrate_rays_kernel<<<nblocks, TPB, 0, stream>>>(rgb, w, idx, out, n, iters);
}